// TopKPooling_32796370273056
// MI455X (gfx1250) — compile-verified
//
#include <hip/hip_runtime.h>
#include <hip/hip_bf16.h>

#define N_NODES 100000
#define N_FEAT  256
#define N_EDGES 3200000
#define K_KEEP  50000
#define PAD     131072          // next pow2 >= N_NODES
#define CHUNK   4096            // elements per LDS sort block (32 KB of u64)

typedef __attribute__((ext_vector_type(2))) float v2f;
typedef __attribute__((ext_vector_type(8))) float v8f;

// ---------------------------------------------------------------------------
// Kernel 1: scores = tanh(X . w) via V_WMMA_F32_16X16X4_F32.
// One wave32 computes 16 node scores. Operand roles chosen for zero masking:
//   A (16x4, SRC0) = weight chunk broadcast to every M row: all lanes read the
//       same sw[k0+2*half .. +1] pair from LDS (uniform broadcast, no EXEC
//       games in the hot loop).
//   B (4x16, SRC1) = x tile: lane L carries node nodeBase+(L&15); VGPR0/1 hold
//       K=0/1 for lanes 0-15 and K=2/3 for lanes 16-31 (ISA B-layout).
//   D row M=0 = the 16 dot products, striped across lanes 0-15 in VGPR0, so
//       the epilogue is one coalesced 16-lane store of tanh(c[0]).
// EXEC is all-1s at every WMMA (wave-uniform early exit only).
// ---------------------------------------------------------------------------
__global__ void tkp_scores_wmma(const float* __restrict__ x,
                                const float* __restrict__ w,
                                float* __restrict__ scores) {
  __shared__ float sw[N_FEAT];
  for (int i = threadIdx.x; i < N_FEAT; i += blockDim.x) sw[i] = w[i];
  __syncthreads();

  const int lane = threadIdx.x & 31;
  const int waveInBlk = threadIdx.x >> 5;
  const int waveId = blockIdx.x * (blockDim.x >> 5) + waveInBlk;
  if (waveId * 16 >= N_NODES) return;         // wave-uniform: EXEC stays all-1s

  const int nodeBase = waveId * 16;
  const int half = lane >> 4;                 // 0: K0/K1 side, 1: K2/K3 side
  const int m    = lane & 15;                 // B column (node), D column (N)
  const float* __restrict__ row = x + (size_t)(nodeBase + m) * N_FEAT;

  v8f c = {};
  #pragma unroll 8
  for (int k0 = 0; k0 < N_FEAT; k0 += 4) {
    v2f a;                                    // weights, broadcast to all rows
    a.x = sw[k0 + 2 * half];
    a.y = sw[k0 + 2 * half + 1];
    v2f b;                                    // x tile, per-node columns
    b.x = row[k0 + 2 * half];
    b.y = row[k0 + 2 * half + 1];
    c = __builtin_amdgcn_wmma_f32_16x16x4_f32(false, a, false, b,
                                              (short)0, c, false, false);
  }
  if (lane < 16) {                            // D[0][n] lives in lane n, VGPR0
    scores[nodeBase + m] = tanhf(c[0]);       // coalesced 64B store per wave
  }
}

// ---------------------------------------------------------------------------
// Kernel 2: build sortable 64-bit keys.
// asc(bits) is the standard monotone float->uint map; key = (~asc)<<32 | idx
// so ascending u64 order == descending score with lowest-index-first ties,
// which exactly matches lax.top_k's stable ordering.
// ---------------------------------------------------------------------------
__global__ void tkp_build_keys(const float* __restrict__ scores,
                               unsigned long long* __restrict__ keys) {
  const int i = blockIdx.x * blockDim.x + threadIdx.x;
  if (i >= PAD) return;
  if (i < N_NODES) {
    unsigned int bits = __float_as_uint(scores[i]);
    unsigned int asc  = (bits & 0x80000000u) ? ~bits : (bits | 0x80000000u);
    keys[i] = ((unsigned long long)(~asc) << 32) | (unsigned int)i;
  } else {
    keys[i] = 0xFFFFFFFFFFFFFFFFull;          // padding sorts last
  }
}

// ---------------------------------------------------------------------------
// Bitonic sort (1 MB of keys -> L2 resident). LDS kernels handle all strides
// <= CHUNK/2; only strides >= CHUNK touch global memory.
// Direction uses the GLOBAL element index bit, so chunk-local passes compose
// correctly into the full-array sort.
// ---------------------------------------------------------------------------
__device__ __forceinline__ void cmpswap(unsigned long long* sk, int i, int j,
                                        bool up) {
  unsigned long long a = sk[i], b = sk[j];
  if ((a > b) == up) { sk[i] = b; sk[j] = a; }
}

__global__ void tkp_bitonic_local_sort(unsigned long long* __restrict__ keys) {
  __shared__ unsigned long long sk[CHUNK];
  const int base = blockIdx.x * CHUNK;
  for (int i = threadIdx.x; i < CHUNK; i += blockDim.x) sk[i] = keys[base + i];
  __syncthreads();
  for (int size = 2; size <= CHUNK; size <<= 1) {
    for (int stride = size >> 1; stride > 0; stride >>= 1) {
      for (int t = threadIdx.x; t < CHUNK / 2; t += blockDim.x) {
        const int i = 2 * t - (t & (stride - 1));   // pow2-stride expansion
        const bool up = (((base + i) & size) == 0);
        cmpswap(sk, i, i + stride, up);
      }
      __syncthreads();
    }
  }
  for (int i = threadIdx.x; i < CHUNK; i += blockDim.x) keys[base + i] = sk[i];
}

__global__ void tkp_bitonic_global_step(unsigned long long* __restrict__ keys,
                                        int size, int stride) {
  const int t = blockIdx.x * blockDim.x + threadIdx.x;   // < PAD/2
  const int i = 2 * t - (t & (stride - 1));
  const int j = i + stride;
  const bool up = ((i & size) == 0);
  unsigned long long a = keys[i], b = keys[j];
  if ((a > b) == up) { keys[i] = b; keys[j] = a; }
}

__global__ void tkp_bitonic_local_merge(unsigned long long* __restrict__ keys,
                                        int size) {
  __shared__ unsigned long long sk[CHUNK];
  const int base = blockIdx.x * CHUNK;
  for (int i = threadIdx.x; i < CHUNK; i += blockDim.x) sk[i] = keys[base + i];
  __syncthreads();
  for (int stride = CHUNK / 2; stride > 0; stride >>= 1) {
    for (int t = threadIdx.x; t < CHUNK / 2; t += blockDim.x) {
      const int i = 2 * t - (t & (stride - 1));
      const bool up = (((base + i) & size) == 0);
      cmpswap(sk, i, i + stride, up);
    }
    __syncthreads();
  }
  for (int i = threadIdx.x; i < CHUNK; i += blockDim.x) keys[base + i] = sk[i];
}

// ---------------------------------------------------------------------------
// Node relabeling table + perm output (as f32).
// ---------------------------------------------------------------------------
__global__ void tkp_init_new_index(int* __restrict__ ni) {
  const int i = blockIdx.x * blockDim.x + threadIdx.x;
  if (i < N_NODES) ni[i] = -1;
}

__global__ void tkp_scatter_perm(const unsigned long long* __restrict__ keys,
                                 int* __restrict__ ni,
                                 float* __restrict__ out_perm) {
  const int i = blockIdx.x * blockDim.x + threadIdx.x;
  if (i >= K_KEEP) return;
  const int node = (int)(keys[i] & 0xFFFFFFFFull);
  ni[node] = i;
  out_perm[i] = (float)node;
}

// ---------------------------------------------------------------------------
// pooled_x[i,:] = x[perm[i],:] * (2 * scores[perm[i]]).
// 256 threads handle 4 rows; each thread moves one float4 (coalesced 1 KB row).
// ---------------------------------------------------------------------------
__global__ void tkp_pooled_x(const float* __restrict__ x,
                             const float* __restrict__ scores,
                             const unsigned long long* __restrict__ keys,
                             float* __restrict__ out) {
  const int row = blockIdx.x * 4 + (threadIdx.x >> 6);
  if (row >= K_KEEP) return;
  const int t = threadIdx.x & 63;                        // 64 * float4 = 256
  const int node = (int)(keys[row] & 0xFFFFFFFFull);
  const float scale = 2.0f * scores[node];
  const float4* __restrict__ src =
      (const float4*)(x + (size_t)node * N_FEAT);
  float4* __restrict__ dst = (float4*)(out + (size_t)row * N_FEAT);
  float4 v = src[t];
  v.x *= scale; v.y *= scale; v.z *= scale; v.w *= scale;
  dst[t] = v;
}

// ---------------------------------------------------------------------------
// Edge relabel. new_index is 400 KB -> L2 resident, gathers are cheap.
// ---------------------------------------------------------------------------
__global__ void tkp_edges(const int* __restrict__ ei,
                          const int* __restrict__ ni,
                          float* __restrict__ out_pe,
                          float* __restrict__ out_mask) {
  const int e = blockIdx.x * blockDim.x + threadIdx.x;
  if (e >= N_EDGES) return;
  __builtin_prefetch(ei + e + 8192, 0, 0);               // global_prefetch_b8
  __builtin_prefetch(ei + N_EDGES + e + 8192, 0, 0);
  const int r = ei[e];
  const int c = ei[N_EDGES + e];
  const int nr = ni[r];
  const int nc = ni[c];
  const bool m = (nr >= 0) & (nc >= 0);
  out_pe[e]           = (float)(m ? nr : -1);
  out_pe[N_EDGES + e] = (float)(m ? nc : -1);
  out_mask[e]         = m ? 1.0f : 0.0f;
}

// ---------------------------------------------------------------------------
extern "C" void kernel_launch(void* const* d_in, const int* in_sizes, int n_in,
                              void* d_out, int out_size, void* d_ws, size_t ws_size,
                              hipStream_t stream) {
  const float* x  = (const float*)d_in[0];     // [100000, 256] f32
  const int*   ei = (const int*)  d_in[1];     // [2, 3200000] i32
  const float* w  = (const float*)d_in[2];     // [1, 256] f32

  // workspace partition
  float*              scores = (float*)d_ws;                                  // 400 KB
  unsigned long long* keys   = (unsigned long long*)((char*)d_ws + (512u << 10)); // 1 MB
  int*                ni     = (int*)((char*)d_ws + (1536u << 10));           // 400 KB

  // output partition (all f32): pooled_x | pooled_edge_index | edge_mask | perm
  float* out      = (float*)d_out;
  float* out_px   = out;                                   // 12,800,000
  float* out_pe   = out + (size_t)K_KEEP * N_FEAT;         //  6,400,000
  float* out_mask = out_pe + 2 * (size_t)N_EDGES;          //  3,200,000
  float* out_perm = out_mask + (size_t)N_EDGES;            //     50,000

  // 1) scores via WMMA: 6250 waves of 16 nodes, 8 waves per 256-thread block
  tkp_scores_wmma<<<782, 256, 0, stream>>>(x, w, scores);

  // 2) sortable keys
  tkp_build_keys<<<PAD / 256, 256, 0, stream>>>(scores, keys);

  // 3) bitonic sort: LDS pass covers sizes 2..4096
  tkp_bitonic_local_sort<<<PAD / CHUNK, 512, 0, stream>>>(keys);
  for (int size = CHUNK * 2; size <= PAD; size <<= 1) {
    for (int stride = size >> 1; stride >= CHUNK; stride >>= 1)
      tkp_bitonic_global_step<<<PAD / 2 / 256, 256, 0, stream>>>(keys, size, stride);
    tkp_bitonic_local_merge<<<PAD / CHUNK, 512, 0, stream>>>(keys, size);
  }

  // 4) relabel table + perm output
  tkp_init_new_index<<<(N_NODES + 255) / 256, 256, 0, stream>>>(ni);
  tkp_scatter_perm<<<(K_KEEP + 255) / 256, 256, 0, stream>>>(keys, ni, out_perm);

  // 5) pooled features (4 rows per 256-thread block)
  tkp_pooled_x<<<K_KEEP / 4, 256, 0, stream>>>(x, scores, keys, out_px);

  // 6) edge relabel + mask
  tkp_edges<<<(N_EDGES + 255) / 256, 256, 0, stream>>>(ei, ni, out_pe, out_mask);
}